// AttentionSimple_22411139351169
// MI455X (gfx1250) — compile-verified
//
#include <hip/hip_runtime.h>
#include <math.h>

#define BATCH 128
#define SEQ   8192
#define EMB   50
#define VOCAB 50000
#define OUTD  2
#define SPLIT 4            // S-dimension split for occupancy (512 blocks total)
#define TPB   256          // 8 wave32s per block
#define CTX_STRIDE 52      // EMB padded to 13 * K=4 WMMA steps
#define PART_STRIDE 56     // [m, l, pad, pad, c[52]]

typedef __attribute__((ext_vector_type(2))) float v2f;
typedef __attribute__((ext_vector_type(8))) float v8f;

// ---------------------------------------------------------------------------
// Kernel 1: gather + online-softmax partials.
// One block per (batch, S-split). Each lane owns tokens s = tid + n*TPB,
// keeps the gathered 50-float row in registers so it is read from L2 once,
// maintains (m, l, c[50]) online, then the block tree-reduces in LDS.
// ---------------------------------------------------------------------------
__global__ __launch_bounds__(TPB) void attn_partial(
    const float* __restrict__ q, const int* __restrict__ kidx,
    const float* __restrict__ emb, float* __restrict__ part)
{
  const int b   = blockIdx.x;
  const int sp  = blockIdx.y;
  const int tid = threadIdx.x;

  // q row in registers (broadcast load, L2/L0 resident)
  float qr[EMB];
  #pragma unroll
  for (int j = 0; j < EMB; ++j) qr[j] = q[b * EMB + j];

  float m = -3.4e38f, l = 0.f;
  float c[EMB];
  #pragma unroll
  for (int j = 0; j < EMB; ++j) c[j] = 0.f;

  const int s0 = sp * (SEQ / SPLIT);
  const int s1 = s0 + (SEQ / SPLIT);
  const int* __restrict__ krow = kidx + (size_t)b * SEQ;

  for (int s = s0 + tid; s < s1; s += TPB) {
    const int idx = krow[s];
    // prefetch next token's embedding row into the cache hierarchy
    const int snext = s + TPB;
    if (snext < s1) {
      const int idxn = krow[snext];
      __builtin_prefetch(emb + (size_t)idxn * EMB, 0, 0);
    }
    // 25 x float2: rows are 200B and idx*50 is even -> 8B aligned loads
    const float2* __restrict__ r2 =
        reinterpret_cast<const float2*>(emb + (size_t)idx * EMB);
    float2 ev[25];
    float dot = 0.f;
    #pragma unroll
    for (int j = 0; j < 25; ++j) {
      ev[j] = r2[j];
      dot = fmaf(ev[j].x, qr[2 * j + 0], dot);
      dot = fmaf(ev[j].y, qr[2 * j + 1], dot);
    }
    const float mn    = fmaxf(m, dot);
    const float scale = __expf(m - mn);
    const float w     = __expf(dot - mn);
    l = fmaf(l, scale, w);
    #pragma unroll
    for (int j = 0; j < 25; ++j) {
      c[2 * j + 0] = fmaf(c[2 * j + 0], scale, w * ev[j].x);
      c[2 * j + 1] = fmaf(c[2 * j + 1], scale, w * ev[j].y);
    }
    m = mn;
  }

  // block tree reduction of (m, l, c[]) in LDS (~55 KB of the 320 KB WGP LDS)
  __shared__ float sm[TPB];
  __shared__ float sl[TPB];
  __shared__ float sc[TPB][CTX_STRIDE];
  sm[tid] = m;
  sl[tid] = l;
  #pragma unroll
  for (int j = 0; j < EMB; ++j) sc[tid][j] = c[j];
  sc[tid][50] = 0.f;
  sc[tid][51] = 0.f;
  __syncthreads();

  for (int stride = TPB / 2; stride > 0; stride >>= 1) {
    if (tid < stride) {
      const float m1 = sm[tid], m2 = sm[tid + stride];
      const float M  = fmaxf(m1, m2);
      const float a1 = __expf(m1 - M);
      const float a2 = __expf(m2 - M);
      sm[tid] = M;
      sl[tid] = sl[tid] * a1 + sl[tid + stride] * a2;
      for (int j = 0; j < EMB; ++j)
        sc[tid][j] = sc[tid][j] * a1 + sc[tid + stride][j] * a2;
    }
    __syncthreads();
  }

  float* __restrict__ p = part + ((size_t)b * SPLIT + sp) * PART_STRIDE;
  if (tid == 0) { p[0] = sm[0]; p[1] = sl[0]; }
  if (tid < CTX_STRIDE) p[4 + tid] = sc[0][tid];
}

// ---------------------------------------------------------------------------
// Kernel 2: merge the SPLIT partials per batch (split-softmax combine),
// normalize, write K-padded ctx rows for the WMMA projection.
// ---------------------------------------------------------------------------
__global__ __launch_bounds__(64) void attn_combine(
    const float* __restrict__ part, float* __restrict__ ctx)
{
  const int b   = blockIdx.x;
  const int tid = threadIdx.x;
  const float* __restrict__ pb = part + (size_t)b * SPLIT * PART_STRIDE;

  float M = -3.4e38f;
  #pragma unroll
  for (int w = 0; w < SPLIT; ++w) M = fmaxf(M, pb[w * PART_STRIDE + 0]);
  float L = 0.f;
  #pragma unroll
  for (int w = 0; w < SPLIT; ++w)
    L += pb[w * PART_STRIDE + 1] * __expf(pb[w * PART_STRIDE + 0] - M);

  if (tid < CTX_STRIDE) {
    float acc = 0.f;
    #pragma unroll
    for (int w = 0; w < SPLIT; ++w)
      acc += pb[w * PART_STRIDE + 4 + tid] * __expf(pb[w * PART_STRIDE + 0] - M);
    ctx[(size_t)b * CTX_STRIDE + tid] = (tid < EMB) ? (acc / L) : 0.f;
  }
}

// ---------------------------------------------------------------------------
// Kernel 3: out[128,2] = ctx[128,50] @ W.T + b via V_WMMA_F32_16X16X4_F32.
// 8 waves, one 16-row M tile each; K padded to 52 -> 13 steps of K=4.
// A frag (16x4 f32, 2 VGPRs): lanes 0-15 hold K={0,1}, lanes 16-31 K={2,3}.
// B frag (4x16 f32, 2 VGPRs): mirrored; only N<2 columns are nonzero.
// D (16x16 f32, 8 VGPRs): VGPR r = row r (lanes 0-15) / row 8+r (lanes 16-31).
// ---------------------------------------------------------------------------
__global__ __launch_bounds__(256) void proj_wmma(
    const float* __restrict__ ctx, const float* __restrict__ W,
    const float* __restrict__ bias, float* __restrict__ out)
{
  const int lane = threadIdx.x & 31;
  const int wave = threadIdx.x >> 5;   // M tile index 0..7
  const int half = lane >> 4;          // 0: K pair {0,1}; 1: K pair {2,3}
  const int ln   = lane & 15;          // A row within tile / B-D column N
  const int rowA = wave * 16 + ln;

  v8f acc = {};
  #pragma unroll
  for (int kk = 0; kk < 13; ++kk) {
    const int k0 = kk * 4 + half * 2;
    v2f a, bv;
    a.x = ctx[(size_t)rowA * CTX_STRIDE + k0 + 0];
    a.y = ctx[(size_t)rowA * CTX_STRIDE + k0 + 1];
    float b0 = 0.f, b1 = 0.f;
    if (ln < OUTD) {                   // B[k][n] = W[n][k], zero-padded
      if (k0 + 0 < EMB) b0 = W[ln * EMB + k0 + 0];
      if (k0 + 1 < EMB) b1 = W[ln * EMB + k0 + 1];
    }
    bv.x = b0; bv.y = b1;
    // 8 args: (neg_a, A, neg_b, B, c_mod, C, reuse_a, reuse_b)
    acc = __builtin_amdgcn_wmma_f32_16x16x4_f32(
        false, a, false, bv, (short)0, acc, false, false);
  }

  if (ln < OUTD) {
    #pragma unroll
    for (int r = 0; r < 8; ++r) {
      const int row = wave * 16 + half * 8 + r;
      out[row * OUTD + ln] = acc[r] + bias[ln];
    }
  }
}

extern "C" void kernel_launch(void* const* d_in, const int* in_sizes, int n_in,
                              void* d_out, int out_size, void* d_ws, size_t ws_size,
                              hipStream_t stream) {
  (void)in_sizes; (void)n_in; (void)out_size; (void)ws_size;

  const float* q    = (const float*)d_in[0];
  const int*   kidx = (const int*)d_in[1];
  const float* emb  = (const float*)d_in[2];
  const float* W    = (const float*)d_in[3];
  const float* bias = (const float*)d_in[4];
  float*       out  = (float*)d_out;

  // ws layout: partials [B*SPLIT*PART_STRIDE] then ctx [B*CTX_STRIDE] (~142 KB)
  float* part = (float*)d_ws;
  float* ctx  = part + (size_t)BATCH * SPLIT * PART_STRIDE;

  dim3 g1(BATCH, SPLIT);
  attn_partial<<<g1, TPB, 0, stream>>>(q, kidx, emb, part);
  attn_combine<<<BATCH, 64, 0, stream>>>(part, ctx);
  proj_wmma<<<1, 256, 0, stream>>>(ctx, W, bias, out);
}